// GridEncoderGeometry_10754598109275
// MI455X (gfx1250) — compile-verified
//
#include <hip/hip_runtime.h>
#include <stdint.h>

#define NUM_LEVELS 16
#define TOTAL_PARAMS 7131240
#define HASH_MASK 0x7FFFFu            // 2^19 - 1 (hashed levels)
#define L0_ELEMS 4920                 // level-0 table entries (dense, r1=17)
#define P1 2654435761u
#define P2 805459861u
#define ENC_BLOCK 1024                // 32 waves: staging amortization + latency hiding

// Per-level base offsets into the table (computed from the reference's
// _compute_offsets(): dense levels 0..2, hashed levels 3..15 of 2^19 each).
__device__ constexpr int LVL_OFF[NUM_LEVELS] = {
    0, 4920, 40864, 315496, 839784, 1364072, 1888360, 2412648,
    2936936, 3461224, 3985512, 4509800, 5034088, 5558376, 6082664, 6606952};

// ---------------------------------------------------------------------------
// Phase 1a: scatter-add embeddings and counts into the table workspace.
// ---------------------------------------------------------------------------
__global__ void k_scatter(const float* __restrict__ emb,
                          const int* __restrict__ sidx,
                          float* __restrict__ sums,
                          uint32_t* __restrict__ counts, int n) {
  int t = blockIdx.x * blockDim.x + threadIdx.x;
  if (t >= n) return;
  int idx = sidx[t];
  float2 e = ((const float2*)emb)[t];
  unsafeAtomicAdd(&sums[2 * idx + 0], e.x);   // global_atomic_add_f32
  unsafeAtomicAdd(&sums[2 * idx + 1], e.y);
  atomicAdd(&counts[idx], 1u);                // global_atomic_add_u32
}

// ---------------------------------------------------------------------------
// Phase 1b: blend (in place over sums):  table = cnt ? sums/cnt : fs
// ---------------------------------------------------------------------------
__global__ void k_blend(float* __restrict__ sums,
                        const uint32_t* __restrict__ counts,
                        const float* __restrict__ fs, int n) {
  int i = blockIdx.x * blockDim.x + threadIdx.x;
  if (i >= n) return;
  uint32_t c = counts[i];
  float2 s = ((const float2*)sums)[i];
  float2 f = ((const float2*)fs)[i];
  float2 r;
  if (c != 0u) {
    float inv = 1.0f / (float)c;   // counts <= 2^24, exact in f32
    r.x = s.x * inv;
    r.y = s.y * inv;
  } else {
    r = f;
  }
  ((float2*)sums)[i] = r;
}

// ---------------------------------------------------------------------------
// Phase 2: multires hash-grid encode. Level-0 table (39.4 KB) is staged into
// LDS with gfx1250 async global->LDS copies; hot gathers then hit
// ds_load_2addr_b64. 1024-thread blocks: 64 KB of level-0 gathers served per
// 39.4 KB staged.
// ---------------------------------------------------------------------------
__global__ void __launch_bounds__(ENC_BLOCK)
k_encode(const float* __restrict__ xin, const float* __restrict__ table,
         const int* __restrict__ bound_p, float* __restrict__ out, int B) {
  __shared__ float sm0[L0_ELEMS * 2];  // 39,360 bytes

  // --- async stage level-0 slice of the table into LDS -------------------
  {
    const uint32_t nchunk = (L0_ELEMS * 2u * 4u) / 16u;  // 2460 x b128
    uint32_t lbase = (uint32_t)(uintptr_t)(&sm0[0]);     // LDS byte offset
    for (uint32_t i = threadIdx.x; i < nchunk; i += blockDim.x) {
      uint32_t loff = lbase + i * 16u;
      uint32_t goff = i * 16u;
      asm volatile("global_load_async_to_lds_b128 %0, %1, %2"
                   :
                   : "v"(loff), "v"(goff), "s"(table)
                   : "memory");
    }
    asm volatile("s_wait_asynccnt 0x0" ::: "memory");
    __syncthreads();
  }

  int b = blockIdx.x * blockDim.x + threadIdx.x;
  if (b >= B) return;

  float bnd = (float)bound_p[0];
  float inv2b = 1.0f / (2.0f * bnd);
  float x = (xin[3 * b + 0] + bnd) * inv2b;
  float y = (xin[3 * b + 1] + bnd) * inv2b;
  float z = (xin[3 * b + 2] + bnd) * inv2b;

  const float2* t2 = (const float2*)table;
  const float2* l2 = (const float2*)sm0;
  float2* o2 = (float2*)out + (size_t)b * NUM_LEVELS;

#pragma unroll
  for (int L = 0; L < NUM_LEVELS; ++L) {
    const float scale = (float)(16 << L) - 1.0f;
    const int off = LVL_OFF[L];

    float px = x * scale + 0.5f;
    float py = y * scale + 0.5f;
    float pz = z * scale + 0.5f;
    float gx = floorf(px), gy = floorf(py), gz = floorf(pz);
    int ix = (int)gx, iy = (int)gy, iz = (int)gz;
    float fx = px - gx, fy = py - gy, fz = pz - gz;
    float wx[2] = {1.0f - fx, fx};
    float wy[2] = {1.0f - fy, fy};
    float wz[2] = {1.0f - fz, fz};

    float o0 = 0.0f, o1 = 0.0f;
#pragma unroll
    for (int c = 0; c < 8; ++c) {
      const int bx = c & 1, by = (c >> 1) & 1, bz = c >> 2;
      uint32_t cx = (uint32_t)(ix + bx);
      uint32_t cy = (uint32_t)(iy + by);
      uint32_t cz = (uint32_t)(iz + bz);
      uint32_t idx;
      if (L < 3) {
        const uint32_t r1 = (16u << L) + 1u;           // 17 / 33 / 65
        idx = cx + cy * r1 + cz * r1 * r1;             // < r1^3 <= hashmap: mod elided
      } else {
        idx = (cx * 1u) ^ (cy * P1) ^ (cz * P2);
        idx &= HASH_MASK;
      }
      float w = wx[bx] * wy[by] * wz[bz];
      float2 v = (L == 0) ? l2[idx] : t2[off + (int)idx];
      o0 = fmaf(w, v.x, o0);
      o1 = fmaf(w, v.y, o1);
    }
    o2[L] = make_float2(o0, o1);
  }
}

// ---------------------------------------------------------------------------
extern "C" void kernel_launch(void* const* d_in, const int* in_sizes, int n_in,
                              void* d_out, int out_size, void* d_ws, size_t ws_size,
                              hipStream_t stream) {
  const float* inputs     = (const float*)d_in[0];
  const float* embeddings = (const float*)d_in[1];
  const float* fs_emb     = (const float*)d_in[2];
  const int*   sidx       = (const int*)d_in[3];
  const int*   bound      = (const int*)d_in[4];

  int B = in_sizes[0] / 3;   // 262144 points
  int N = in_sizes[1] / 2;   // 2,000,000 scatter points

  // Workspace layout: [sums/table: TOTAL_PARAMS*2 f32][counts: TOTAL_PARAMS u32]
  float*    sums   = (float*)d_ws;
  uint32_t* counts = (uint32_t*)((char*)d_ws + (size_t)TOTAL_PARAMS * 2 * sizeof(float));
  size_t zbytes = (size_t)TOTAL_PARAMS * 2 * sizeof(float) +
                  (size_t)TOTAL_PARAMS * sizeof(uint32_t);
  hipMemsetAsync(d_ws, 0, zbytes, stream);

  k_scatter<<<(N + 255) / 256, 256, 0, stream>>>(embeddings, sidx, sums, counts, N);
  k_blend<<<(TOTAL_PARAMS + 255) / 256, 256, 0, stream>>>(sums, counts, fs_emb,
                                                          TOTAL_PARAMS);
  k_encode<<<(B + ENC_BLOCK - 1) / ENC_BLOCK, ENC_BLOCK, 0, stream>>>(
      inputs, sums, bound, (float*)d_out, B);
}